// MPNNModel_65094524339280
// MI455X (gfx1250) — compile-verified
//
#include <hip/hip_runtime.h>

// ---------------------------------------------------------------------------
// MPNN (message-passing GNN) on gfx1250 using V_WMMA_F32_16X16X4_F32.
//   msg  = (relu([x_i|x_j|dpos] @ mw1 + mb1)) @ mw2 + mb2        per edge
//   aggr = scatter_mean(msg, i)                                   per node
//   out  = (relu([x|aggr] @ aw1 + ab1)) @ aw2 + ab2               per node
// Weights are repacked once into K-pair-interleaved float2 layout
// (zero-padded K) so every B fragment is one unconditional global b64 load.
// ---------------------------------------------------------------------------

typedef __attribute__((ext_vector_type(2))) float v2f;
typedef __attribute__((ext_vector_type(4))) float v4f;
typedef __attribute__((ext_vector_type(8))) float v8f;

#define N_NODES 100000
#define N_EDGES 1600000
#define D_FEAT  64
#define D_HID   64

// packed-weight sizes (in float2 elements)
#define WP1_PAIRS 66   // K: 131 -> 132 -> 66 pairs
#define WP2_PAIRS 32   // K: 64  -> 32 pairs
#define WA1_PAIRS 64   // K: 128 -> 64 pairs
#define WA2_PAIRS 32   // K: 64  -> 32 pairs

// Same-wave LDS store->load ordering (LDS pipeline is in-order per wave on
// CDNA5; the memory clobber stops the compiler reordering across it).
__device__ __forceinline__ void lds_fence_wave() {
  asm volatile("s_wait_dscnt 0" ::: "memory");
}

__device__ __forceinline__ v8f wmma_f32(v2f a, v2f b, v8f c) {
  return __builtin_amdgcn_wmma_f32_16x16x4_f32(
      false, a, false, b, (short)0, c, false, false);
}

__device__ __forceinline__ v8f splat8(float v) {
  v8f r;
#pragma unroll
  for (int i = 0; i < 8; ++i) r[i] = v;
  return r;
}

// ---------------------------------------------------------------------------
// Repack weights: wp[p][n] = { W[2p][n], W[2p+1][n] }, zero-padded K rows.
// ---------------------------------------------------------------------------
__global__ void pack_weights_kernel(const float* __restrict__ mw1,
                                    const float* __restrict__ mw2,
                                    const float* __restrict__ aw1,
                                    const float* __restrict__ aw2,
                                    float* __restrict__ wp1,
                                    float* __restrict__ wp2,
                                    float* __restrict__ wa1,
                                    float* __restrict__ wa2) {
  const int tid    = blockIdx.x * blockDim.x + threadIdx.x;
  const int stride = gridDim.x * blockDim.x;

  for (int idx = tid; idx < WP1_PAIRS * 64; idx += stride) {
    const int p = idx >> 6, n = idx & 63;
    const int k0 = 2 * p, k1 = 2 * p + 1;
    wp1[idx * 2 + 0] = (k0 < 131) ? mw1[k0 * 64 + n] : 0.0f;
    wp1[idx * 2 + 1] = (k1 < 131) ? mw1[k1 * 64 + n] : 0.0f;
  }
  for (int idx = tid; idx < WP2_PAIRS * 64; idx += stride) {
    const int p = idx >> 6, n = idx & 63;
    wp2[idx * 2 + 0] = mw2[(2 * p) * 64 + n];
    wp2[idx * 2 + 1] = mw2[(2 * p + 1) * 64 + n];
  }
  for (int idx = tid; idx < WA1_PAIRS * 64; idx += stride) {
    const int p = idx >> 6, n = idx & 63;
    wa1[idx * 2 + 0] = aw1[(2 * p) * 64 + n];
    wa1[idx * 2 + 1] = aw1[(2 * p + 1) * 64 + n];
  }
  for (int idx = tid; idx < WA2_PAIRS * 64; idx += stride) {
    const int p = idx >> 6, n = idx & 63;
    wa2[idx * 2 + 0] = aw2[(2 * p) * 64 + n];
    wa2[idx * 2 + 1] = aw2[(2 * p + 1) * 64 + n];
  }
}

// ---------------------------------------------------------------------------
// Edge kernel: 64 threads = 2 waves; each wave owns a 32-edge supertile
// (two 16-row WMMA tiles sharing B fragments).
// ---------------------------------------------------------------------------
__global__ __launch_bounds__(64)
void mpnn_edge_kernel(const float* __restrict__ x,
                      const int*   __restrict__ ei,      // [2][N_EDGES]
                      const float* __restrict__ pos,     // [N_NODES][3]
                      const v2f*   __restrict__ wp1,     // packed [66][64]
                      const float* __restrict__ mb1,
                      const v2f*   __restrict__ wp2,     // packed [32][64]
                      const float* __restrict__ mb2,
                      float* __restrict__ sum,           // [N_NODES][64]
                      float* __restrict__ cnt)           // [N_NODES]
{
  __shared__ float A[2][32][132];   // [wave][edge-row][132 feat]
  __shared__ float H[2][32][66];    // hidden tile, even stride
  __shared__ int   IDX[2][32];

  const int wave = threadIdx.x >> 5;
  const int lane = threadIdx.x & 31;
  const int half = lane >> 4;       // 0: lanes 0-15, 1: lanes 16-31
  const int ml   = lane & 15;       // M-row (A frag) / N-col (B,C frags)

  float (*Aw)[132] = A[wave];
  float (*Hw)[66]  = H[wave];
  int*   idxw      = IDX[wave];

  // bias fragments (loop-invariant)
  float b1v[4], b2v[4];
#pragma unroll
  for (int t = 0; t < 4; ++t) {
    b1v[t] = mb1[t * 16 + ml];
    b2v[t] = mb2[t * 16 + ml];
  }

  const int nSuper = N_EDGES / 32;  // 50000, exact
  const int waveId = blockIdx.x * 2 + wave;
  const int nWaves = gridDim.x * 2;

  for (int sp = waveId; sp < nSuper; sp += nWaves) {
    const int e0 = sp * 32;

    // ---- gather [x_i | x_j | dpos | 0]: one edge row per lane ----
    {
      const int e  = e0 + lane;
      const int ii = ei[e];
      const int jj = ei[N_EDGES + e];
      idxw[lane]   = ii;
      const float* xi = x + (size_t)ii * D_FEAT;
      const float* xj = x + (size_t)jj * D_FEAT;
      float* dst = &Aw[lane][0];
#pragma unroll
      for (int c = 0; c < D_FEAT; c += 4) {
        *(v4f*)(dst + c)          = *(const v4f*)(xi + c);
        *(v4f*)(dst + D_FEAT + c) = *(const v4f*)(xj + c);
      }
#pragma unroll
      for (int c = 0; c < 3; ++c)
        dst[128 + c] = pos[(size_t)jj * 3 + c] - pos[(size_t)ii * 3 + c];
      dst[131] = 0.0f;
    }
    lds_fence_wave();

    // ---- GEMM1: [32x132] @ mw1 (+mb1), K zero-padded in packed weights ----
    v8f acc0[4], acc1[4];
#pragma unroll
    for (int t = 0; t < 4; ++t) {
      acc0[t] = splat8(b1v[t]);
      acc1[t] = splat8(b1v[t]);
    }

#pragma unroll 3
    for (int k0 = 0; k0 < 132; k0 += 4) {
      const int k2 = (k0 >> 1) + half;
      const int ka = k0 + 2 * half;
      v2f a0 = *(const v2f*)&Aw[ml][ka];
      v2f a1 = *(const v2f*)&Aw[16 + ml][ka];
#pragma unroll
      for (int t = 0; t < 4; ++t) {
        v2f b = wp1[k2 * 64 + t * 16 + ml];
        acc0[t] = wmma_f32(a0, b, acc0[t]);
        acc1[t] = wmma_f32(a1, b, acc1[t]);
      }
    }

    // ---- ReLU, stage hidden tiles ----
#pragma unroll
    for (int t = 0; t < 4; ++t)
#pragma unroll
      for (int r = 0; r < 8; ++r) {
        const int m = r + 8 * half;
        float v0 = acc0[t][r], v1 = acc1[t][r];
        Hw[m][t * 16 + ml]      = v0 > 0.0f ? v0 : 0.0f;
        Hw[16 + m][t * 16 + ml] = v1 > 0.0f ? v1 : 0.0f;
      }
    lds_fence_wave();

    // ---- GEMM2: H[32x64] @ mw2 (+mb2) ----
    v8f m0[4], m1[4];
#pragma unroll
    for (int t = 0; t < 4; ++t) {
      m0[t] = splat8(b2v[t]);
      m1[t] = splat8(b2v[t]);
    }

#pragma unroll 4
    for (int k0 = 0; k0 < 64; k0 += 4) {
      const int k2 = (k0 >> 1) + half;
      const int ka = k0 + 2 * half;
      v2f a0 = *(const v2f*)&Hw[ml][ka];
      v2f a1 = *(const v2f*)&Hw[16 + ml][ka];
#pragma unroll
      for (int t = 0; t < 4; ++t) {
        v2f b = wp2[k2 * 64 + t * 16 + ml];
        m0[t] = wmma_f32(a0, b, m0[t]);
        m1[t] = wmma_f32(a1, b, m1[t]);
      }
    }

    // ---- scatter-add messages + counts ----
#pragma unroll
    for (int t = 0; t < 4; ++t)
#pragma unroll
      for (int r = 0; r < 8; ++r) {
        const int m = r + 8 * half;
        const int n = t * 16 + ml;
        atomicAdd(&sum[(size_t)idxw[m] * D_HID + n],      m0[t][r]);
        atomicAdd(&sum[(size_t)idxw[16 + m] * D_HID + n], m1[t][r]);
      }
    atomicAdd(&cnt[idxw[lane]], 1.0f);
  }
}

// ---------------------------------------------------------------------------
// Node kernel: u = [x | sum/max(cnt,1)], 32-node supertile per wave.
// ---------------------------------------------------------------------------
__global__ __launch_bounds__(64)
void mpnn_node_kernel(const float* __restrict__ x,
                      const float* __restrict__ sum,
                      const float* __restrict__ cnt,
                      const v2f*   __restrict__ wa1,   // packed [64][64]
                      const float* __restrict__ ab1,
                      const v2f*   __restrict__ wa2,   // packed [32][64]
                      const float* __restrict__ ab2,
                      float* __restrict__ out)         // [N_NODES][64]
{
  __shared__ float U[2][32][132];
  __shared__ float H[2][32][66];

  const int wave = threadIdx.x >> 5;
  const int lane = threadIdx.x & 31;
  const int half = lane >> 4;
  const int ml   = lane & 15;

  float (*Uw)[132] = U[wave];
  float (*Hw)[66]  = H[wave];

  float b1v[4], b2v[4];
#pragma unroll
  for (int t = 0; t < 4; ++t) {
    b1v[t] = ab1[t * 16 + ml];
    b2v[t] = ab2[t * 16 + ml];
  }

  const int nSuper = N_NODES / 32;  // 3125, exact
  const int waveId = blockIdx.x * 2 + wave;
  const int nWaves = gridDim.x * 2;

  for (int sp = waveId; sp < nSuper; sp += nWaves) {
    const int n0 = sp * 32;

    // ---- build [x | mean-aggr]: one node row per lane ----
    {
      const size_t node = (size_t)(n0 + lane);
      const float  c    = cnt[node];
      const float  inv  = 1.0f / (c > 1.0f ? c : 1.0f);
      const float* xv = x   + node * D_FEAT;
      const float* sv = sum + node * D_HID;
      float* dst = &Uw[lane][0];
#pragma unroll
      for (int cc = 0; cc < D_FEAT; cc += 4) {
        *(v4f*)(dst + cc) = *(const v4f*)(xv + cc);
        v4f s = *(const v4f*)(sv + cc);
        v4f m; m.x = s.x * inv; m.y = s.y * inv; m.z = s.z * inv; m.w = s.w * inv;
        *(v4f*)(dst + D_FEAT + cc) = m;
      }
    }
    lds_fence_wave();

    // ---- GEMM1: U[32x128] @ aw1 (+ab1) ----
    v8f acc0[4], acc1[4];
#pragma unroll
    for (int t = 0; t < 4; ++t) {
      acc0[t] = splat8(b1v[t]);
      acc1[t] = splat8(b1v[t]);
    }

#pragma unroll 4
    for (int k0 = 0; k0 < 128; k0 += 4) {
      const int k2 = (k0 >> 1) + half;
      const int ka = k0 + 2 * half;
      v2f a0 = *(const v2f*)&Uw[ml][ka];
      v2f a1 = *(const v2f*)&Uw[16 + ml][ka];
#pragma unroll
      for (int t = 0; t < 4; ++t) {
        v2f b = wa1[k2 * 64 + t * 16 + ml];
        acc0[t] = wmma_f32(a0, b, acc0[t]);
        acc1[t] = wmma_f32(a1, b, acc1[t]);
      }
    }

#pragma unroll
    for (int t = 0; t < 4; ++t)
#pragma unroll
      for (int r = 0; r < 8; ++r) {
        const int m = r + 8 * half;
        float v0 = acc0[t][r], v1 = acc1[t][r];
        Hw[m][t * 16 + ml]      = v0 > 0.0f ? v0 : 0.0f;
        Hw[16 + m][t * 16 + ml] = v1 > 0.0f ? v1 : 0.0f;
      }
    lds_fence_wave();

    // ---- GEMM2: H[32x64] @ aw2 (+ab2) ----
    v8f o0[4], o1[4];
#pragma unroll
    for (int t = 0; t < 4; ++t) {
      o0[t] = splat8(b2v[t]);
      o1[t] = splat8(b2v[t]);
    }

#pragma unroll 4
    for (int k0 = 0; k0 < 64; k0 += 4) {
      const int k2 = (k0 >> 1) + half;
      const int ka = k0 + 2 * half;
      v2f a0 = *(const v2f*)&Hw[ml][ka];
      v2f a1 = *(const v2f*)&Hw[16 + ml][ka];
#pragma unroll
      for (int t = 0; t < 4; ++t) {
        v2f b = wa2[k2 * 64 + t * 16 + ml];
        o0[t] = wmma_f32(a0, b, o0[t]);
        o1[t] = wmma_f32(a1, b, o1[t]);
      }
    }

    // ---- store ----
#pragma unroll
    for (int t = 0; t < 4; ++t)
#pragma unroll
      for (int r = 0; r < 8; ++r) {
        const int m = r + 8 * half;
        const int n = t * 16 + ml;
        out[(size_t)(n0 + m) * D_HID + n]      = o0[t][r];
        out[(size_t)(n0 + 16 + m) * D_HID + n] = o1[t][r];
      }
  }
}

// ---------------------------------------------------------------------------
extern "C" void kernel_launch(void* const* d_in, const int* in_sizes, int n_in,
                              void* d_out, int out_size, void* d_ws, size_t ws_size,
                              hipStream_t stream) {
  const float* x   = (const float*)d_in[0];
  const int*   ei  = (const int*)  d_in[1];
  const float* pos = (const float*)d_in[2];
  const float* mw1 = (const float*)d_in[3];
  const float* mb1 = (const float*)d_in[4];
  const float* mw2 = (const float*)d_in[5];
  const float* mb2 = (const float*)d_in[6];
  const float* aw1 = (const float*)d_in[7];
  const float* ab1 = (const float*)d_in[8];
  const float* aw2 = (const float*)d_in[9];
  const float* ab2 = (const float*)d_in[10];

  float* out = (float*)d_out;

  // workspace layout (floats)
  float* sum = (float*)d_ws;                          // [N_NODES][64]
  float* cnt = sum + (size_t)N_NODES * D_HID;         // [N_NODES]
  float* wp1 = cnt + N_NODES;                         // 66*64 float2
  float* wp2 = wp1 + WP1_PAIRS * 64 * 2;              // 32*64 float2
  float* wa1 = wp2 + WP2_PAIRS * 64 * 2;              // 64*64 float2
  float* wa2 = wa1 + WA1_PAIRS * 64 * 2;              // 32*64 float2

  (void)in_sizes; (void)n_in; (void)out_size; (void)ws_size;

  hipMemsetAsync(d_ws, 0,
                 ((size_t)N_NODES * D_HID + N_NODES) * sizeof(float), stream);

  pack_weights_kernel<<<64, 256, 0, stream>>>(mw1, mw2, aw1, aw2,
                                              wp1, wp2, wa1, wa2);

  mpnn_edge_kernel<<<4096, 64, 0, stream>>>(x, ei, pos,
                                            (const v2f*)wp1, mb1,
                                            (const v2f*)wp2, mb2,
                                            sum, cnt);
  mpnn_node_kernel<<<1024, 64, 0, stream>>>(x, sum, cnt,
                                            (const v2f*)wa1, ab1,
                                            (const v2f*)wa2, ab2,
                                            out);
}